// MessagePassingLayer_75703093559398
// MI455X (gfx1250) — compile-verified
//
#include <hip/hip_runtime.h>
#include <math.h>

// Problem constants (from reference): B=4, N=256, F=128, H=256, STEPS=6
#define Bn 4
#define Nn 256
#define Fn 128
#define Hn 256
#define Mrows (Bn * Nn)   // 1024

typedef float v2f __attribute__((ext_vector_type(2)));
typedef float v8f __attribute__((ext_vector_type(8)));

// ---------------------------------------------------------------------------
// Generic f32 WMMA GEMM:  out[m][n] = sum_k in[m*lda+k] * W[n*ldw+wofs+k]
//                         (+ rowscale[m]*bias[n]  if bias != nullptr)
// One wave computes one 16x16 tile using V_WMMA_F32_16X16X4_F32.
// A 16x4 fragment : lanes 0-15 = M rows, VGPR0/1 = K0/K1; lanes 16-31 = K2/K3
// B 4x16 fragment : lanes 0-15 = N cols, VGPR0/1 = K0/K1; lanes 16-31 = K2/K3
// D 16x16        : VGPR r = row (r + 8*hi), col = lane&15
// ---------------------------------------------------------------------------
__global__ __launch_bounds__(256) void gemm_wmma_f32(
    const float* __restrict__ in, const float* __restrict__ W,
    float* __restrict__ out, int M, int Nout, int K,
    int lda, int ldw, int wofs, int ldo,
    const float* __restrict__ bias, const float* __restrict__ rowscale)
{
    const int lane = threadIdx.x & 31;
    const int wave = threadIdx.x >> 5;
    const int ntn  = Nout >> 4;
    const int tile = blockIdx.x * (blockDim.x >> 5) + wave;
    const int total = (M >> 4) * ntn;
    if (tile >= total) return;            // wave-uniform exit: EXEC stays all-1

    const int mt = tile / ntn;
    const int nt = tile - mt * ntn;
    const int m0 = mt << 4;
    const int n0 = nt << 4;
    const int hi  = lane >> 4;            // 0: lanes 0-15, 1: lanes 16-31
    const int l15 = lane & 15;

    const float* __restrict__ arow = in + (size_t)(m0 + l15) * lda + (hi << 1);
    const float* __restrict__ wrow = W  + (size_t)(n0 + l15) * ldw + wofs + (hi << 1);

    v8f acc = {};
    #pragma unroll 8
    for (int k = 0; k < K; k += 4) {
        v2f av, bv;
        av.x = arow[k];  av.y = arow[k + 1];
        bv.x = wrow[k];  bv.y = wrow[k + 1];
        acc = __builtin_amdgcn_wmma_f32_16x16x4_f32(
            false, av, false, bv, (short)0, acc, false, false);
    }

    const float bn = bias ? bias[n0 + l15] : 0.0f;
    #pragma unroll
    for (int r = 0; r < 8; ++r) {
        const int m = m0 + (hi << 3) + r;
        float v = acc[r];
        if (bias) v += (rowscale ? rowscale[m] : 1.0f) * bn;
        out[(size_t)m * ldo + n0 + l15] = v;
    }
}

// ---------------------------------------------------------------------------
// Message aggregation (factored):
//   s[b,i,k]  = sum_j valid[b,i,j] * relu(a[b,i,k] + c[b,j,k] + adj[b,i,j]*w1e[k] + b1[k])
//   cnt[b,i]  = sum_j valid[b,i,j]
// valid = (adj>0) & (mask_i>0) & (mask_j>0).  One block per (b,i); thread = k.
// ---------------------------------------------------------------------------
__global__ __launch_bounds__(Hn) void msg_agg_kernel(
    const float* __restrict__ a, const float* __restrict__ c,
    const float* __restrict__ adj, const float* __restrict__ mask,
    const float* __restrict__ W1, const float* __restrict__ b1,
    float* __restrict__ s, float* __restrict__ cnt)
{
    __shared__ float s_adj[Nn];
    __shared__ float s_mj[Nn];

    const int bi = blockIdx.x;            // b*N + i
    const int b  = bi >> 8;
    const int t  = threadIdx.x;           // channel k (0..255)

    s_adj[t] = adj[(size_t)bi * Nn + t];  // row i of adj for this (b,i)
    s_mj[t]  = mask[(b << 8) + t];
    __syncthreads();

    const float ai = a[(size_t)bi * Hn + t];
    const float we = W1[(size_t)t * (2 * Fn + 1) + 2 * Fn]; // w1e[t] = W1[t][256]
    const float bb = b1[t];
    const float mi = mask[bi];

    const float* __restrict__ crow = c + ((size_t)(b << 8) * Hn) + t;

    float acc = 0.0f, cn = 0.0f;
    #pragma unroll 4
    for (int j = 0; j < Nn; ++j) {
        const float ad = s_adj[j];
        const bool  v  = (ad > 0.0f) & (s_mj[j] > 0.0f) & (mi > 0.0f);
        const float pre = ai + crow[(size_t)j * Hn] + ad * we + bb;
        if (v) { acc += fmaxf(pre, 0.0f); cn += 1.0f; }
    }
    s[(size_t)bi * Hn + t] = acc;
    if (t == 0) cnt[bi] = cn;
}

// ---------------------------------------------------------------------------
// GRU cell update (per node, per feature):
//   r = sig(gi[:,f]    + gh[:,f])
//   z = sig(gi[:,F+f]  + gh[:,F+f])
//   n = tanh(gi[:,2F+f]+ r*gh[:,2F+f])
//   h' = ((1-z)*n + z*h) * mask
// ---------------------------------------------------------------------------
__global__ __launch_bounds__(256) void gru_kernel(
    const float* __restrict__ gi, const float* __restrict__ gh,
    const float* __restrict__ hin, const float* __restrict__ mask,
    float* __restrict__ hout)
{
    const int idx = blockIdx.x * blockDim.x + threadIdx.x;  // < M*F
    if (idx >= Mrows * Fn) return;
    const int m = idx >> 7;
    const int f = idx & (Fn - 1);
    const float* gim = gi + (size_t)m * (3 * Fn);
    const float* ghm = gh + (size_t)m * (3 * Fn);

    const float r = 1.0f / (1.0f + expf(-(gim[f] + ghm[f])));
    const float z = 1.0f / (1.0f + expf(-(gim[Fn + f] + ghm[Fn + f])));
    const float n = tanhf(gim[2 * Fn + f] + r * ghm[2 * Fn + f]);
    const float h = hin[idx];
    hout[idx] = ((1.0f - z) * n + z * h) * mask[m];
}

// ---------------------------------------------------------------------------
// Host-side orchestration
// ---------------------------------------------------------------------------
extern "C" void kernel_launch(void* const* d_in, const int* in_sizes, int n_in,
                              void* d_out, int out_size, void* d_ws, size_t ws_size,
                              hipStream_t stream)
{
    (void)in_sizes; (void)n_in; (void)out_size; (void)ws_size;

    const float* x    = (const float*)d_in[0];   // (B,N,F)
    const float* adj  = (const float*)d_in[1];   // (B,N,N)
    const float* mask = (const float*)d_in[2];   // (B,N)
    const float* W1   = (const float*)d_in[3];   // (H, 2F+1) row-major, ld=257
    const float* b1   = (const float*)d_in[4];   // (H,)
    const float* W2   = (const float*)d_in[5];   // (H,H)
    const float* b2   = (const float*)d_in[6];   // (H,)
    const float* Wih  = (const float*)d_in[7];   // (3F,H)
    const float* bih  = (const float*)d_in[8];   // (3F,)
    const float* Whh  = (const float*)d_in[9];   // (3F,F)
    const float* bhh  = (const float*)d_in[10];  // (3F,)

    float* ws = (float*)d_ws;
    float* a_buf = ws;                 ws += Mrows * Hn;       // 1024*256
    float* c_buf = ws;                 ws += Mrows * Hn;
    float* s_buf = ws;                 ws += Mrows * Hn;
    float* agg   = ws;                 ws += Mrows * Hn;
    float* gi    = ws;                 ws += Mrows * 3 * Fn;   // 1024*384
    float* gh    = ws;                 ws += Mrows * 3 * Fn;
    float* cnt   = ws;                 ws += Mrows;
    float* hA    = ws;                 ws += Mrows * Fn;       // 1024*128
    float* hB    = ws;                 ws += Mrows * Fn;

    auto gemm_blocks = [](int M, int Nout) {
        const int tiles = (M >> 4) * (Nout >> 4);
        return (tiles + 7) / 8;        // 8 waves (16x16 tiles) per 256-thread block
    };

    float* hout_seq[6] = { hA, hB, hA, hB, hA, (float*)d_out };
    const float* hcur = x;

    for (int step = 0; step < 6; ++step) {
        // a = h @ W1i^T   (K=F=128, W1 cols [0,128), ldw=257)
        gemm_wmma_f32<<<gemm_blocks(Mrows, Hn), 256, 0, stream>>>(
            hcur, W1, a_buf, Mrows, Hn, Fn, Fn, 2 * Fn + 1, 0, Hn, nullptr, nullptr);
        // c = h @ W1j^T   (W1 cols [128,256))
        gemm_wmma_f32<<<gemm_blocks(Mrows, Hn), 256, 0, stream>>>(
            hcur, W1, c_buf, Mrows, Hn, Fn, Fn, 2 * Fn + 1, Fn, Hn, nullptr, nullptr);

        // s[b,i,k], cnt[b,i]  (factored masked-relu reduction over j)
        msg_agg_kernel<<<Mrows, Hn, 0, stream>>>(
            a_buf, c_buf, adj, mask, W1, b1, s_buf, cnt);

        // agg = s @ W2^T + cnt*b2   (K=H=256)
        gemm_wmma_f32<<<gemm_blocks(Mrows, Hn), 256, 0, stream>>>(
            s_buf, W2, agg, Mrows, Hn, Hn, Hn, Hn, 0, Hn, b2, cnt);

        // gi = agg @ Wih^T + bih    (Nout=384, K=256)
        gemm_wmma_f32<<<gemm_blocks(Mrows, 3 * Fn), 256, 0, stream>>>(
            agg, Wih, gi, Mrows, 3 * Fn, Hn, Hn, Hn, 0, 3 * Fn, bih, nullptr);

        // gh = h @ Whh^T + bhh      (Nout=384, K=128)
        gemm_wmma_f32<<<gemm_blocks(Mrows, 3 * Fn), 256, 0, stream>>>(
            hcur, Whh, gh, Mrows, 3 * Fn, Fn, Fn, Fn, 0, 3 * Fn, bhh, nullptr);

        // GRU update + mask
        gru_kernel<<<(Mrows * Fn + 255) / 256, 256, 0, stream>>>(
            gi, gh, hcur, mask, hout_seq[step]);

        hcur = hout_seq[step];
    }
}